// ACLIP_80075370266823
// MI455X (gfx1250) — compile-verified
//
#include <hip/hip_runtime.h>

typedef float v2f __attribute__((ext_vector_type(2)));
typedef float v8f __attribute__((ext_vector_type(8)));

#define NB   256
#define NTOK 577
#define DIM  1024
#define NPAT 576
#define KEEP 288

__launch_bounds__(256)
__global__ void aclip_fused_kernel(const float* __restrict__ img,
                                   const float* __restrict__ txt,
                                   float* __restrict__ out)
{
    __shared__ float          ldsT[DIM];       // raw text vector for this batch
    __shared__ float          ldsSim[NPAT];    // dot(patch, t)
    __shared__ float          ldsSsq[NPAT];    // ||patch||^2
    __shared__ unsigned short ldsRank[NPAT];
    __shared__ unsigned short ldsKeep[KEEP];

    const int b    = blockIdx.x;
    const int tid  = threadIdx.x;
    const int lane = tid & 31;
    const int wave = tid >> 5;
    const int half = lane >> 4;    // 0: lanes 0-15, 1: lanes 16-31
    const int l    = lane & 15;

    // ---- Phase 0: stage text vector in LDS (1024 floats = 256 float4) ----
    {
        const float4* t4 = (const float4*)(txt + (size_t)b * DIM);
        ((float4*)ldsT)[tid] = t4[tid];
    }
    __syncthreads();

    const size_t imgBase = (size_t)b * NTOK * DIM;

    // ---- Phase 1: sims via v_wmma_f32_16x16x4_f32, 16 patches per wave-tile ----
    for (int tile = wave; tile < NPAT / 16; tile += 8) {
        const int p0 = tile * 16;
        // A layout (ISA 16x4 f32): lane l holds row p0+l; half=0 -> K+{0,1}, half=1 -> K+{2,3}
        const float* rowp = img + imgBase + (size_t)(1 + p0 + l) * DIM + 2 * half;
        // B layout (4x16 f32, broadcast cols): lanes<16 hold t[K],t[K+1]; lanes>=16 t[K+2],t[K+3]
        const float* tp = ldsT + 2 * half;

        v8f c0 = {}; v8f c1 = {}; v8f c2 = {}; v8f c3 = {};
        float s0 = 0.f, s1 = 0.f, s2 = 0.f, s3 = 0.f;

        for (int k = 0; k < DIM; k += 16) {
            v2f a0 = *(const v2f*)(rowp + k);
            v2f a1 = *(const v2f*)(rowp + k + 4);
            v2f a2 = *(const v2f*)(rowp + k + 8);
            v2f a3 = *(const v2f*)(rowp + k + 12);
            v2f b0 = *(const v2f*)(tp + k);
            v2f b1 = *(const v2f*)(tp + k + 4);
            v2f b2 = *(const v2f*)(tp + k + 8);
            v2f b3 = *(const v2f*)(tp + k + 12);
            c0 = __builtin_amdgcn_wmma_f32_16x16x4_f32(false, a0, false, b0, (short)0, c0, false, false);
            c1 = __builtin_amdgcn_wmma_f32_16x16x4_f32(false, a1, false, b1, (short)0, c1, false, false);
            c2 = __builtin_amdgcn_wmma_f32_16x16x4_f32(false, a2, false, b2, (short)0, c2, false, false);
            c3 = __builtin_amdgcn_wmma_f32_16x16x4_f32(false, a3, false, b3, (short)0, c3, false, false);
            s0 = fmaf(a0.x, a0.x, s0); s0 = fmaf(a0.y, a0.y, s0);
            s1 = fmaf(a1.x, a1.x, s1); s1 = fmaf(a1.y, a1.y, s1);
            s2 = fmaf(a2.x, a2.x, s2); s2 = fmaf(a2.y, a2.y, s2);
            s3 = fmaf(a3.x, a3.x, s3); s3 = fmaf(a3.y, a3.y, s3);
        }
        v8f   c = (c0 + c1) + (c2 + c3);
        float s = (s0 + s1) + (s2 + s3);

        // combine K-halves of sum-of-squares: lanes l and l+16 both end with row p0+l ssq
        s += __shfl_xor(s, 16, 32);
        if (half == 0) ldsSsq[p0 + l] = s;

        // Broadcast-B makes all D columns identical: lane n VGPR r = sim[p0 + r (+8 if lane>=16)]
        if (l == 0) {
            const int base = p0 + 8 * half;
            #pragma unroll
            for (int j = 0; j < 8; ++j) ldsSim[base + j] = c[j];
        }
    }
    __syncthreads();

    // ---- Phase 2: scale by 1/max(||patch||, eps)  (||text|| is ranking-invariant) ----
    for (int p = tid; p < NPAT; p += 256) {
        float nrm = fmaxf(sqrtf(ldsSsq[p]), 1e-12f);
        ldsSim[p] = ldsSim[p] / nrm;
    }
    __syncthreads();

    // ---- Phase 3: stable rank (ties -> lower index wins, matching lax.top_k) ----
    for (int p = tid; p < NPAT; p += 256) {
        const float my = ldsSim[p];
        int cnt = 0;
        for (int q = 0; q < NPAT; ++q) {
            const float sq = ldsSim[q];
            cnt += (sq > my) || (sq == my && q < p);
        }
        ldsRank[p] = (unsigned short)cnt;
    }
    __syncthreads();

    // ---- Phase 4: compact kept indices in ascending order ----
    for (int p = tid; p < NPAT; p += 256) {
        if (ldsRank[p] < KEEP) {
            int pos = 0;
            for (int q = 0; q < p; ++q) pos += (ldsRank[q] < KEEP);
            ldsKeep[pos] = (unsigned short)p;
        }
    }
    __syncthreads();

    // ---- Phase 5: gather rows with float4 (b128) traffic ----
    float* outB = out + (size_t)b * (1 + KEEP) * DIM;
    {   // cls token -> out row 0
        const float4* s4 = (const float4*)(img + imgBase);
        ((float4*)outB)[tid] = s4[tid];
    }
    for (int j = 0; j < KEEP; ++j) {
        const int p = ldsKeep[j];
        const float4* s4 = (const float4*)(img + imgBase + (size_t)(1 + p) * DIM);
        float4* d4 = (float4*)(outB + (size_t)(1 + j) * DIM);
        d4[tid] = s4[tid];
    }
}

extern "C" void kernel_launch(void* const* d_in, const int* in_sizes, int n_in,
                              void* d_out, int out_size, void* d_ws, size_t ws_size,
                              hipStream_t stream) {
    (void)in_sizes; (void)n_in; (void)out_size; (void)d_ws; (void)ws_size;
    const float* img = (const float*)d_in[0];   // (256, 577, 1024) fp32
    const float* txt = (const float*)d_in[1];   // (256, 1024) fp32
    float* out = (float*)d_out;                 // (256, 289, 1024) fp32
    aclip_fused_kernel<<<NB, 256, 0, stream>>>(img, txt, out);
}